// SummationMPNN_57423712748201
// MI455X (gfx1250) — compile-verified
//
#include <hip/hip_runtime.h>
#include <math.h>

typedef __attribute__((ext_vector_type(2))) float v2f;
typedef __attribute__((ext_vector_type(8))) float v8f;

#define BATCH   32
#define NNODE   128
#define HIDF    100
#define HP      112          // 7 * 16, padded hidden width
#define EDGEF   4
#define OUTF    100
#define ROWS    (BATCH * NNODE)   // 4096
#define NT_H    (HP / 16)         // 7 col tiles
#define MT_R    (ROWS / 16)       // 256 row tiles
#define K2P     208               // 13 * 16, padded concat width

// ---------------------------------------------------------------- WMMA helper
__device__ __forceinline__ v8f wmma4(v2f a, v2f b, v8f c) {
  // D = A(16x4 f32) * B(4x16 f32) + C ; full f32 — matches reference numerics
  return __builtin_amdgcn_wmma_f32_16x16x4_f32(
      /*neg_a=*/false, a, /*neg_b=*/false, b,
      /*c_mod=*/(short)0, c, /*reuse_a=*/false, /*reuse_b=*/false);
}

// ---------------------------------------------------------------- setup kernels
__global__ void pad_kernel(const float* __restrict__ src, float* __restrict__ dst,
                           int R, int C, int RP, int CP) {
  int i = blockIdx.x * blockDim.x + threadIdx.x;
  if (i >= RP * CP) return;
  int r = i / CP, c = i % CP;
  dst[i] = (r < R && c < C) ? src[r * C + c] : 0.0f;
}

__global__ void mask_kernel(const float* __restrict__ edges, float* __restrict__ nmask) {
  int row = blockIdx.x * blockDim.x + threadIdx.x;  // row = b*N + v
  if (row >= ROWS) return;
  const float* p = edges + (size_t)row * NNODE * EDGEF;
  float s = 0.0f;
  for (int i = 0; i < NNODE * EDGEF; ++i) s += p[i];
  nmask[row] = (s != 0.0f) ? 1.0f : 0.0f;
}

__global__ void zero_kernel(float* __restrict__ p, int n) {
  int i = blockIdx.x * blockDim.x + threadIdx.x;
  if (i < n) p[i] = 0.0f;
}

// ---------------------------------------------------------------- hv/hw GEMM
// hv = hidden @ Wv_pad ; hw = hidden @ Ww_pad    ([4096,112] x [112,112])
// One wave -> one 16x16 tile of BOTH outputs (shared A operand, 2 accumulators).
__global__ void __launch_bounds__(256)
gemm_hv_hw(const float* __restrict__ hid, const float* __restrict__ Wvp,
           const float* __restrict__ Wwp, float* __restrict__ hv,
           float* __restrict__ hw) {
  int wid = blockIdx.x * (blockDim.x >> 5) + (threadIdx.x >> 5);
  if (wid >= MT_R * NT_H) return;
  int mt = wid / NT_H, nt = wid % NT_H;

  int lane = threadIdx.x & 31;
  int half = lane >> 4;       // 0 or 1
  int l16  = lane & 15;
  int rowA = mt * 16 + l16;   // A row this lane supplies
  int col  = nt * 16 + l16;   // B/D column this lane supplies

  v8f accV = {}, accW = {};
  for (int k0 = 0; k0 < HP; k0 += 4) {
    int k = k0 + 2 * half;
    v2f a, bv, bw;
    a.x  = hid[rowA * HP + k];
    a.y  = hid[rowA * HP + k + 1];
    bv.x = Wvp[k * HP + col];
    bv.y = Wvp[(k + 1) * HP + col];
    bw.x = Wwp[k * HP + col];
    bw.y = Wwp[(k + 1) * HP + col];
    accV = wmma4(a, bv, accV);          // A reused for both products
    accW = wmma4(a, bw, accW);
  }
#pragma unroll
  for (int r = 0; r < 8; ++r) {
    int row = mt * 16 + r + 8 * half;
    hv[row * HP + col] = accV[r];
    hw[row * HP + col] = accW[r];
  }
}

// ---------------------------------------------------------------- message pass
// messages[b,v,h] = sum_w edge_mask * relu(hv[b,v,h] + hw[b,w,h] + edges[b,v,w,:]@We)
// One wave per (b,v); all 8 waves of a block share batch b, so hw[b,:,:] (56KB)
// is staged once per block into LDS. e_proj tile computed on the fly (K=4 WMMA).
__global__ void __launch_bounds__(256)
msg_kernel(const float* __restrict__ edges, const float* __restrict__ Wep,
           const float* __restrict__ hv, const float* __restrict__ hw,
           float* __restrict__ msgs) {
  __shared__ float hws[NNODE * HP];           // 56 KB: hw slab for this block's b
  __shared__ float smask[8][NNODE];           // 4 KB : per-wave edge-mask row
  int tid   = threadIdx.x;
  int wband = tid >> 5;
  int wid   = blockIdx.x * 8 + wband;         // wid = b*N + v (grid is exact)
  int lane  = tid & 31;
  int half  = lane >> 4;
  int l16   = lane & 15;
  int bbase = (wid >> 7) << 7;                // b * N (same for whole block)

  const float* erow = edges + (size_t)wid * NNODE * EDGEF;
  __builtin_prefetch(erow, 0, 1);             // global_prefetch_b8 of edge row

  // cooperative b128 stage of hw[b,:,:] into LDS (all 256 threads)
  {
    const float4* src = (const float4*)(hw + (size_t)bbase * HP);
    float4* dst = (float4*)hws;
    for (int i = tid; i < (NNODE * HP) / 4; i += 256) dst[i] = src[i];
  }

  // edge_mask[b,v,w] for all 128 neighbors (wave-private row)
  for (int w = lane; w < NNODE; w += 32) {
    const float* e = erow + w * EDGEF;
    float s = e[0] + e[1] + e[2] + e[3];
    smask[wband][w] = (s != 0.0f) ? 1.0f : 0.0f;
  }
  __syncthreads();

  for (int ht = 0; ht < NT_H; ++ht) {
    int hc = ht * 16 + l16;                   // output column (h)
    float hvv = hv[wid * HP + hc];
    // B operand constant across the w loop — hoisted
    v2f b;
    b.x = Wep[(2 * half) * HP + hc];
    b.y = Wep[(2 * half + 1) * HP + hc];
    float msum = 0.0f;
    for (int wt = 0; wt < NNODE / 16; ++wt) { // 8 w tiles
      int wA = wt * 16 + l16;                 // A row (neighbor w)
      v2f a;
      a.x = erow[wA * EDGEF + 2 * half];
      a.y = erow[wA * EDGEF + 2 * half + 1];
      v8f acc = {};
      acc = wmma4(a, b, acc);                 // e_proj 16x16 tile (K=4 exact)
      float part = 0.0f;
#pragma unroll
      for (int r = 0; r < 8; ++r) {
        int w = wt * 16 + r + 8 * half;       // D row = neighbor index
        float val = acc[r] + hvv + hws[w * HP + hc];   // LDS read, not global
        val = val > 0.0f ? val : 0.0f;        // relu
        part += val * smask[wband][w];        // edge mask
      }
      msum += part;
    }
    msum += __shfl_xor(msum, 16, 32);         // fold the two lane halves (w groups)
    if (half == 0) msgs[wid * HP + hc] = msum;
  }
}

// ---------------------------------------------------------------- node update
// hid_out = node_mask ? tanh(concat[hid, msgs] @ Wu_pad) : hid
__global__ void __launch_bounds__(256)
update_kernel(const float* __restrict__ hid, const float* __restrict__ msgs,
              const float* __restrict__ Wup, const float* __restrict__ nmask,
              float* __restrict__ hid_out) {
  int wid = blockIdx.x * (blockDim.x >> 5) + (threadIdx.x >> 5);
  if (wid >= MT_R * NT_H) return;
  int mt = wid / NT_H, nt = wid % NT_H;
  int lane = threadIdx.x & 31;
  int half = lane >> 4;
  int l16  = lane & 15;
  int rowA = mt * 16 + l16;
  int col  = nt * 16 + l16;

  v8f acc = {};
  for (int k0 = 0; k0 < K2P; k0 += 4) {
    int k = k0 + 2 * half;
    v2f a, b;
    a.x = (k < HIDF)       ? hid[rowA * HP + k]
        : (k < 2 * HIDF)   ? msgs[rowA * HP + (k - HIDF)] : 0.0f;
    int k1 = k + 1;
    a.y = (k1 < HIDF)      ? hid[rowA * HP + k1]
        : (k1 < 2 * HIDF)  ? msgs[rowA * HP + (k1 - HIDF)] : 0.0f;
    b.x = Wup[k * HP + col];
    b.y = Wup[k1 * HP + col];
    acc = wmma4(a, b, acc);
  }
#pragma unroll
  for (int r = 0; r < 8; ++r) {
    int row = mt * 16 + r + 8 * half;
    float m = nmask[row];
    float upd = tanhf(acc[r]);
    float old = hid[row * HP + col];
    hid_out[row * HP + col] = (m != 0.0f) ? upd : old;
  }
}

// ---------------------------------------------------------------- readout
// out[b,:] += node_mask * relu(concat[hid, nodes] @ Wr_pad)  summed over v
__global__ void __launch_bounds__(256)
readout_kernel(const float* __restrict__ hid, const float* __restrict__ nodespad,
               const float* __restrict__ Wrp, const float* __restrict__ nmask,
               float* __restrict__ out) {
  int wid = blockIdx.x * (blockDim.x >> 5) + (threadIdx.x >> 5);
  if (wid >= MT_R * NT_H) return;
  int mt = wid / NT_H, nt = wid % NT_H;
  int lane = threadIdx.x & 31;
  int half = lane >> 4;
  int l16  = lane & 15;
  int rowA = mt * 16 + l16;
  int col  = nt * 16 + l16;

  v8f acc = {};
  for (int k0 = 0; k0 < K2P; k0 += 4) {
    int k = k0 + 2 * half;
    v2f a, b;
    a.x = (k < HIDF)       ? hid[rowA * HP + k]
        : (k < 2 * HIDF)   ? nodespad[rowA * HP + (k - HIDF)] : 0.0f;
    int k1 = k + 1;
    a.y = (k1 < HIDF)      ? hid[rowA * HP + k1]
        : (k1 < 2 * HIDF)  ? nodespad[rowA * HP + (k1 - HIDF)] : 0.0f;
    b.x = Wrp[k * HP + col];
    b.y = Wrp[k1 * HP + col];
    acc = wmma4(a, b, acc);
  }
  float colsum = 0.0f;
#pragma unroll
  for (int r = 0; r < 8; ++r) {
    int row = mt * 16 + r + 8 * half;
    float v = acc[r] > 0.0f ? acc[r] : 0.0f;        // relu
    colsum += (nmask[row] != 0.0f) ? v : 0.0f;      // node mask
  }
  colsum += __shfl_xor(colsum, 16, 32);             // fold row halves
  int hcol = nt * 16 + l16;
  if (half == 0 && hcol < OUTF) {
    int b = (mt * 16) >> 7;                         // 16 rows share one batch (128%16==0)
    atomicAdd(&out[b * OUTF + hcol], colsum);
  }
}

// ---------------------------------------------------------------- launch
extern "C" void kernel_launch(void* const* d_in, const int* in_sizes, int n_in,
                              void* d_out, int out_size, void* d_ws, size_t ws_size,
                              hipStream_t stream) {
  const float* nodes = (const float*)d_in[0];
  const float* edges = (const float*)d_in[1];
  const float* Wv    = (const float*)d_in[2];
  const float* Ww    = (const float*)d_in[3];
  const float* We    = (const float*)d_in[4];
  const float* Wu    = (const float*)d_in[5];
  const float* Wr    = (const float*)d_in[6];
  float* out = (float*)d_out;

  float* ws = (float*)d_ws;
  size_t off = 0;
  auto alloc = [&](size_t n) { float* p = ws + off; off += n; return p; };
  float* nodespad = alloc((size_t)ROWS * HP);
  float* hidA     = alloc((size_t)ROWS * HP);
  float* hidB     = alloc((size_t)ROWS * HP);
  float* hv       = alloc((size_t)ROWS * HP);
  float* hw       = alloc((size_t)ROWS * HP);
  float* msgs     = alloc((size_t)ROWS * HP);
  float* Wvp      = alloc((size_t)HP * HP);
  float* Wwp      = alloc((size_t)HP * HP);
  float* Wep      = alloc((size_t)EDGEF * HP);
  float* Wup      = alloc((size_t)K2P * HP);
  float* Wrp      = alloc((size_t)K2P * HP);
  float* nmask    = alloc((size_t)ROWS);

  const int T = 256;
  auto blk = [](int n) { return (n + 255) / 256; };

  // weight padding (zero pads guarantee padded K/N columns contribute nothing)
  pad_kernel<<<blk(HP * HP), T, 0, stream>>>(Wv, Wvp, HIDF, HIDF, HP, HP);
  pad_kernel<<<blk(HP * HP), T, 0, stream>>>(Ww, Wwp, HIDF, HIDF, HP, HP);
  pad_kernel<<<blk(EDGEF * HP), T, 0, stream>>>(We, Wep, EDGEF, HIDF, EDGEF, HP);
  pad_kernel<<<blk(K2P * HP), T, 0, stream>>>(Wu, Wup, 2 * HIDF, HIDF, K2P, HP);
  pad_kernel<<<blk(K2P * HP), T, 0, stream>>>(Wr, Wrp, 2 * HIDF, OUTF, K2P, HP);
  // nodes -> padded nodes copy and initial hidden state
  pad_kernel<<<blk(ROWS * HP), T, 0, stream>>>(nodes, nodespad, ROWS, HIDF, ROWS, HP);
  pad_kernel<<<blk(ROWS * HP), T, 0, stream>>>(nodes, hidA, ROWS, HIDF, ROWS, HP);
  mask_kernel<<<blk(ROWS), T, 0, stream>>>(edges, nmask);
  zero_kernel<<<blk(BATCH * OUTF), T, 0, stream>>>(out, BATCH * OUTF);

  const int tile_blocks = (MT_R * NT_H + 7) / 8;   // 224 (16x16 tile per wave)
  const int msg_blocks  = ROWS / 8;                // 512 (one (b,v) per wave)

  float* hin = hidA;
  float* hout = hidB;
  for (int pass = 0; pass < 3; ++pass) {
    gemm_hv_hw<<<tile_blocks, T, 0, stream>>>(hin, Wvp, Wwp, hv, hw);
    msg_kernel<<<msg_blocks, T, 0, stream>>>(edges, Wep, hv, hw, msgs);
    update_kernel<<<tile_blocks, T, 0, stream>>>(hin, msgs, Wup, nmask, hout);
    float* tmp = hin; hin = hout; hout = tmp;
  }
  // after 3 passes final hidden is in `hin` (A->B->A->B)
  readout_kernel<<<tile_blocks, T, 0, stream>>>(hin, nodespad, Wrp, nmask, out);
}